// DecoderTreeNN_55740085567675
// MI455X (gfx1250) — compile-verified
//
#include <hip/hip_runtime.h>

typedef float v2f __attribute__((ext_vector_type(2)));
typedef float v8f __attribute__((ext_vector_type(8)));

#define EMBED_D 128
#define CHUNK 2048
#define WAVES_PER_BLOCK 16          // 8 column tiles x 2 tables (key/val)
#define BLOCK_T (WAVES_PER_BLOCK * 32)

// ---------------------------------------------------------------------------
// Zero-fill the output (d_out is poisoned before timing).
// ---------------------------------------------------------------------------
__global__ void zero_out_kernel(float4* __restrict__ out, int n4) {
    int i = blockIdx.x * blockDim.x + threadIdx.x;
    int stride = gridDim.x * blockDim.x;
    float4 z = make_float4(0.f, 0.f, 0.f, 0.f);
    for (; i < n4; i += stride) out[i] = z;
}

// ---------------------------------------------------------------------------
// Gather + segmented sum via fp32 WMMA (V_WMMA_F32_16X16X4_F32).
//   Out[t][seg][d] = sum over tokens with tree_ids==seg of C_t[token][d]
// Each wave owns a 16-wide column tile of one table and keeps a 16-segment
// x 16-column fp32 accumulator in the WMMA C operand. A = one-hot segment
// selector for 4 tokens, B = gathered embedding slice for those 4 tokens.
// Sorted tree_ids => the 16-row window slides rarely; flushes use HW f32
// atomics so chunk-boundary segments shared between blocks are correct.
// ---------------------------------------------------------------------------
__global__ __launch_bounds__(BLOCK_T)
void seg_embed_sum_kernel(const int* __restrict__ token_ids,
                          const int* __restrict__ tree_ids,
                          const float* __restrict__ C_hop,
                          const float* __restrict__ C_hop1,
                          const int* __restrict__ p_batch,
                          const int* __restrict__ p_mtrees,
                          float* __restrict__ out,
                          int n_tokens) {
    __shared__ int s_tok[CHUNK];
    __shared__ int s_seg[CHUNK];

    const int n_seg = p_batch[0] * p_mtrees[0];
    const int start = blockIdx.x * CHUNK;
    int count = n_tokens - start;
    if (count <= 0) return;
    if (count > CHUNK) count = CHUNK;

    // Stage this chunk's ids into LDS once (coalesced), then all 16 waves
    // read them via cheap ds loads instead of redundant VMEM gathers.
    for (int i = threadIdx.x; i < count; i += BLOCK_T) {
        s_tok[i] = token_ids[start + i];
        s_seg[i] = tree_ids[start + i];
    }
    __syncthreads();

    const int wave = threadIdx.x >> 5;
    const int lane = threadIdx.x & 31;
    const int half = lane >> 4;     // 0: K=0,1  1: K=2,3 (A/B half-wave split)
    const int nidx = lane & 15;     // M row (A) / N column (B, C/D)
    const int table = wave >> 3;    // 0 -> C_hop (keys), 1 -> C_hop1 (values)
    const int col = ((wave & 7) << 4) + nidx;

    const float* __restrict__ tab = table ? C_hop1 : C_hop;
    float* __restrict__ outt = out + (size_t)table * (size_t)n_seg * EMBED_D;

    v8f acc = {0.f, 0.f, 0.f, 0.f, 0.f, 0.f, 0.f, 0.f};
    int base = s_seg[0];            // first segment of current 16-row window

    for (int i = 0; i < count; i += 4) {
        // Group-uniform segment ids (sorted: s0 <= s1 <= s2 <= s3).
        int s0 = s_seg[i];
        int s1 = (i + 1 < count) ? s_seg[i + 1] : s0;
        int s2 = (i + 2 < count) ? s_seg[i + 2] : s1;
        int s3 = (i + 3 < count) ? s_seg[i + 3] : s2;
        int gmax = s3;

        // This lane supplies K slots (half*2, half*2+1) of A and B.
        int ka = i + half * 2;
        int kb = ka + 1;
        bool va = (ka < count);
        bool vb = (kb < count);
        int sa = half ? s2 : s0;
        int sb = half ? s3 : s1;
        int ta = va ? s_tok[ka] : 0;    // PAD row (all zeros) for tail slots
        int tb = vb ? s_tok[kb] : 0;    // keeps B finite -> one-hot 0 is safe

        // B[k][n] = tab[token_k][col]; 16 lanes cover one contiguous 64B run.
        v2f bm;
        bm.x = tab[(size_t)ta * EMBED_D + col];
        bm.y = tab[(size_t)tb * EMBED_D + col];

        while (true) {
            // A one-hot: A[m][k] = (seg_k == base + m); out-of-window -> 0.
            v2f am;
            am.x = (va && sa == base + nidx) ? 1.0f : 0.0f;
            am.y = (vb && sb == base + nidx) ? 1.0f : 0.0f;
            acc = __builtin_amdgcn_wmma_f32_16x16x4_f32(
                false, am, false, bm, (short)0, acc, false, false);
            if (gmax < base + 16) break;

            // Window overflow: flush 16x16 accumulator tile and re-base.
#pragma unroll
            for (int r = 0; r < 8; ++r) {
                int seg = base + r + half * 8;   // C/D: M = r (+8 for lanes 16..31)
                if (seg < n_seg)
                    unsafeAtomicAdd(&outt[(size_t)seg * EMBED_D + col], acc[r]);
            }
            acc = (v8f){0.f, 0.f, 0.f, 0.f, 0.f, 0.f, 0.f, 0.f};
            int nb = base + 16;
            if      (s0 >= nb) base = s0;
            else if (s1 >= nb) base = s1;
            else if (s2 >= nb) base = s2;
            else               base = s3;
            // Re-run WMMA: tokens already accumulated have seg < new base and
            // mask to zero; remaining tokens now land in the new window.
        }
    }

    // Final flush of the open window.
#pragma unroll
    for (int r = 0; r < 8; ++r) {
        int seg = base + r + half * 8;
        if (seg < n_seg)
            unsafeAtomicAdd(&outt[(size_t)seg * EMBED_D + col], acc[r]);
    }
}

// ---------------------------------------------------------------------------
extern "C" void kernel_launch(void* const* d_in, const int* in_sizes, int n_in,
                              void* d_out, int out_size, void* d_ws, size_t ws_size,
                              hipStream_t stream) {
    const int*   token_ids = (const int*)d_in[0];
    const int*   tree_ids  = (const int*)d_in[1];
    const float* C_hop     = (const float*)d_in[2];
    const float* C_hop1    = (const float*)d_in[3];
    const int*   p_batch   = (const int*)d_in[4];
    const int*   p_mtrees  = (const int*)d_in[5];
    float*       out       = (float*)d_out;

    const int n_tokens = in_sizes[0];

    // 1) zero the output (poisoned by harness; atomics accumulate into it)
    int n4 = out_size >> 2;   // out_size = 2*B*max_trees*128, divisible by 4
    int zb = (n4 + 255) / 256;
    if (zb > 4096) zb = 4096;
    zero_out_kernel<<<zb, 256, 0, stream>>>((float4*)out, n4);

    // 2) WMMA segmented gather-sum
    int blocks = (n_tokens + CHUNK - 1) / CHUNK;
    seg_embed_sum_kernel<<<blocks, BLOCK_T, 0, stream>>>(
        token_ids, tree_ids, C_hop, C_hop1, p_batch, p_mtrees, out, n_tokens);
}